// MEF_SSIM_54597624267418
// MI455X (gfx1250) — compile-verified
//
#include <hip/hip_runtime.h>
#include <math.h>

// ---------------------------------------------------------------------------
// MEF-SSIM for MI455X (gfx1250, wave32).
//
// Shapes: LDRseq (B=16, N=3, H=512, W=512, C=3) f32, Fused (16,512,512,3) f32.
// Box filter = 8x8 SAME avg pool with padding-excluded counts. Separable:
//   boxavg(f)[h][w] = S2d(f)[h][w] / (cnt_h(h) * cnt_w(w)),
//   window along each axis = [i-3, i+4] clipped (XLA SAME: pad_lo=3, pad_hi=4).
// mean-over-C commutes with the (linear) box sum, so we filter 11 scalar
// fields per batch: per n {u=mean_c X, v=mean_c X^2, xy=mean_c X*Y} + {p,q}.
//
// k1 : per (b,h) row — fields + horizontal sliding sums (LDS), write hsum;
//      deterministic tree-reduce row sum of u for gMu.
// kg : reduce row partials -> gMu[b][n].
// k2 : per 16x16 tile, 1 wave — vertical sliding sums via chained
//      V_WMMA_F32_16X16X4_F32. Border clipping is folded into the banded
//      A-matrix weights (zero column == zero data row), so all global loads
//      are unconditional from row-clamped addresses (no exec-mask branches).
// k3 : reduce Q partials -> Q[b].
// ---------------------------------------------------------------------------

typedef float v2f __attribute__((ext_vector_type(2)));
typedef float v8f __attribute__((ext_vector_type(8)));

#define HH 512
#define WW 512
#define BB 16
#define NN 3
#define NFIELD 11           // 0..8 = n*3 + {u,v,xy}, 9 = p, 10 = q

static constexpr size_t HS_FLOATS    = (size_t)BB * NFIELD * HH * WW;   // 46,137,344
static constexpr size_t GPART_FLOATS = (size_t)BB * NN * HH;            // 24,576
static constexpr size_t GMU_FLOATS   = (size_t)BB * NN;                 // 48

static constexpr size_t OFF_QSEQ = (size_t)BB * HH * WW;                // 4,194,304
static constexpr size_t OFF_Q    = OFF_QSEQ + (size_t)BB * HH * WW * NN;// 16,777,216

__device__ __forceinline__ int box_cnt(int i) {
  int lo = i - 3; if (lo < 0) lo = 0;
  int hi = i + 4; if (hi > 511) hi = 511;
  return hi - lo + 1;
}

// ---------------- k1: channel means + horizontal sliding sums ----------------
__global__ __launch_bounds__(512) void k1_rowsum(
    const float* __restrict__ X,   // (B,N,H,W,C)
    const float* __restrict__ Y,   // (B,H,W,C)
    float* __restrict__ hs,        // (B,11,H,W)
    float* __restrict__ gpart)     // (B,N,H) row sums of u
{
  const int w = threadIdx.x;
  const int h = blockIdx.x;
  const int b = blockIdx.y;
  __shared__ float bu[WW], bv[WW], bw[WW];

  const float third = 1.0f / 3.0f;
  const size_t yoff = (((size_t)b * HH + h) * WW + w) * 3;
  const float y0 = Y[yoff], y1 = Y[yoff + 1], y2 = Y[yoff + 2];

  // Fused fields p, q
  bu[w] = (y0 + y1 + y2) * third;
  bv[w] = (y0 * y0 + y1 * y1 + y2 * y2) * third;
  __syncthreads();

  int lo = w - 3; if (lo < 0) lo = 0;
  int hi = w + 4; if (hi > 511) hi = 511;

  {
    float sp = 0.f, sq = 0.f;
    for (int k = lo; k <= hi; ++k) { sp += bu[k]; sq += bv[k]; }
    const size_t fb = (size_t)b * NFIELD;
    hs[((fb + 9)  * HH + h) * WW + w] = sp;
    hs[((fb + 10) * HH + h) * WW + w] = sq;
  }

  for (int n = 0; n < NN; ++n) {
    __syncthreads();   // previous-phase LDS reads complete before overwrite
    const size_t xoff = ((((size_t)b * NN + n) * HH + h) * WW + w) * 3;
    const float x0 = X[xoff], x1 = X[xoff + 1], x2 = X[xoff + 2];
    const float u  = (x0 + x1 + x2) * third;
    const float v  = (x0 * x0 + x1 * x1 + x2 * x2) * third;
    const float xy = (x0 * y0 + x1 * y1 + x2 * y2) * third;
    bu[w] = u; bv[w] = v; bw[w] = xy;
    __syncthreads();

    float su = 0.f, sv = 0.f, sw = 0.f;
    for (int k = lo; k <= hi; ++k) { su += bu[k]; sv += bv[k]; sw += bw[k]; }
    const size_t fb = (size_t)b * NFIELD + n * 3;
    hs[((fb + 0) * HH + h) * WW + w] = su;
    hs[((fb + 1) * HH + h) * WW + w] = sv;
    hs[((fb + 2) * HH + h) * WW + w] = sw;
    __syncthreads();

    // deterministic tree reduction of u over the row (for gMu)
    for (int s = 256; s > 0; s >>= 1) {
      if (w < s) bu[w] += bu[w + s];
      __syncthreads();
    }
    if (w == 0) gpart[((size_t)b * NN + n) * HH + h] = bu[0];
  }
}

// ---------------- kg: gMu = mean over H,W,C ----------------
__global__ __launch_bounds__(128) void kg_gmu(const float* __restrict__ gpart,
                                              float* __restrict__ gmu)
{
  const int id = blockIdx.x;        // b*3 + n, 48 blocks
  const int t  = threadIdx.x;
  __shared__ float s[128];
  float v = 0.f;
  for (int i = t; i < HH; i += 128) v += gpart[(size_t)id * HH + i];
  s[t] = v; __syncthreads();
  for (int r = 64; r > 0; r >>= 1) {
    if (t < r) s[t] += s[t + r];
    __syncthreads();
  }
  if (t == 0) gmu[id] = s[0] * (1.0f / ((float)HH * (float)WW));
}

// ---------------- k2: WMMA vertical box sums + SSIM math ----------------
__global__ __launch_bounds__(32) void k2_wmma_ssim(
    const float* __restrict__ hs,   // (B,11,H,W) horizontal sums
    const float* __restrict__ gmu,  // (B,N)
    float* __restrict__ out,        // qmap | qmap_seq | Q
    float* __restrict__ qpart)      // (B,1024) tile partial sums of qmap
{
  const int lane = threadIdx.x;         // 0..31
  const int bx = blockIdx.x;            // column tile (32)
  const int by = blockIdx.y;            // row tile (32)
  const int b  = blockIdx.z;
  const int nl = lane & 15;
  const bool hiHalf = lane >= 16;
  const int col = bx * 16 + nl;
  const int h0 = by * 16 - 3;           // first (virtual) input row of the tile

  // A matrix (16x23 banded ones, chunked K=4) with border clipping folded in:
  //   A[M][k] = (M <= k <= M+7) && (0 <= h0+k < 512).
  // Zeroing column k of A == zeroing data row k of B, so B loads can be
  // unconditional from row-clamped addresses (valid finite data * 0 weight).
  // f32 16x4 A layout: lane L holds M=L%16; VGPR0: K=(L<16?0:2), VGPR1: K=(L<16?1:3).
  v2f aw[6];
  int idx0[6], idx1[6];                 // element index within one field
  const int M = nl;
#pragma unroll
  for (int c = 0; c < 6; ++c) {
    const int k0 = 4 * c + (hiHalf ? 2 : 0);
    const int k1 = 4 * c + (hiHalf ? 3 : 1);
    const int r0 = h0 + k0, r1 = h0 + k1;
    const bool v0 = (r0 >= 0) & (r0 < HH);
    const bool v1 = (r1 >= 0) & (r1 < HH);
    aw[c].x = (k0 >= M && k0 <= M + 7 && v0) ? 1.0f : 0.0f;
    aw[c].y = (k1 >= M && k1 <= M + 7 && v1) ? 1.0f : 0.0f;
    const int cr0 = r0 < 0 ? 0 : (r0 > HH - 1 ? HH - 1 : r0);
    const int cr1 = r1 < 0 ? 0 : (r1 > HH - 1 ? HH - 1 : r1);
    idx0[c] = cr0 * WW + col;
    idx1[c] = cr1 * WW + col;
  }

  v8f S[NFIELD];
#pragma unroll
  for (int f = 0; f < NFIELD; ++f) {
    const float* __restrict__ base = hs + ((size_t)b * NFIELD + f) * HH * WW;
    // 12 unconditional loads, then 6 chained WMMAs
    v2f bv0, bv1, bv2, bv3, bv4, bv5;
    bv0.x = base[idx0[0]]; bv0.y = base[idx1[0]];
    bv1.x = base[idx0[1]]; bv1.y = base[idx1[1]];
    bv2.x = base[idx0[2]]; bv2.y = base[idx1[2]];
    bv3.x = base[idx0[3]]; bv3.y = base[idx1[3]];
    bv4.x = base[idx0[4]]; bv4.y = base[idx1[4]];
    bv5.x = base[idx0[5]]; bv5.y = base[idx1[5]];
    v8f acc = {};
    acc = __builtin_amdgcn_wmma_f32_16x16x4_f32(false, aw[0], false, bv0, (short)0, acc, false, false);
    acc = __builtin_amdgcn_wmma_f32_16x16x4_f32(false, aw[1], false, bv1, (short)0, acc, false, false);
    acc = __builtin_amdgcn_wmma_f32_16x16x4_f32(false, aw[2], false, bv2, (short)0, acc, false, false);
    acc = __builtin_amdgcn_wmma_f32_16x16x4_f32(false, aw[3], false, bv3, (short)0, acc, false, false);
    acc = __builtin_amdgcn_wmma_f32_16x16x4_f32(false, aw[4], false, bv4, (short)0, acc, false, false);
    acc = __builtin_amdgcn_wmma_f32_16x16x4_f32(false, aw[5], false, bv5, (short)0, acc, false, false);
    S[f] = acc;
  }

  // global-mean Gaussian weights (uniform per wave)
  float Lg[NN];
#pragma unroll
  for (int n = 0; n < NN; ++n) {
    const float g = gmu[b * NN + n] - 0.5f;
    Lg[n] = expf(-g * g * 12.5f);       // 1/(2*0.2^2) = 12.5
  }

  const float C1c = 1e-4f;              // (0.01)^2
  const float C2c = 9e-4f;              // (0.03)^2
  const float cw = (float)box_cnt(col);
  float qsum = 0.f;

  // D layout: VGPR r, lanes 0-15: (M=r, N=lane); lanes 16-31: (M=8+r, N=lane-16).
#pragma unroll
  for (int r = 0; r < 8; ++r) {
    const int row = by * 16 + (hiHalf ? 8 + r : r);
    const float icnt = 1.0f / ((float)box_cnt(row) * cw);

    const float muY = S[9][r] * icnt;
    const float sy  = S[10][r] * icnt - muY * muY;

    float muXn[NN], sxn[NN], qn[NN];
    float lxsum = 0.f, muxw_num = 0.f;
#pragma unroll
    for (int n = 0; n < NN; ++n) {
      const float mu = S[n * 3 + 0][r] * icnt;
      muXn[n] = mu;
      sxn[n]  = S[n * 3 + 1][r] * icnt - mu * mu;
      const float d = mu - 0.5f;
      const float lx = expf(-d * d * 12.5f) * Lg[n];
      lxsum += lx;
      muxw_num += mu * lx;
    }
    const float muxw = muxw_num / lxsum;
    const float A1 = 2.0f * muxw * muY + C1c;
    const float B1 = muxw * muxw + muY * muY + C1c;
    const float A1dB1 = A1 / B1;

#pragma unroll
    for (int n = 0; n < NN; ++n) {
      const float sxy = S[n * 3 + 2][r] * icnt - muXn[n] * muY;
      const float A2 = 2.0f * sxy + C2c;
      const float B2 = sxn[n] + sy + C2c;
      qn[n] = A1dB1 * (A2 / B2);
    }

    // argmax over n of sigmaX^2 (first max wins, strict >)
    int best = 0;
    float bs = sxn[0];
    if (sxn[1] > bs) { bs = sxn[1]; best = 1; }
    if (sxn[2] > bs) { bs = sxn[2]; best = 2; }
    const float qsel = qn[best];

    const size_t pix = ((size_t)b * HH + row) * WW + col;
    out[pix] = qsel;
#pragma unroll
    for (int n = 0; n < NN; ++n) out[OFF_QSEQ + pix * 3 + n] = qn[n];
    qsum += qsel;
  }

  // wave32 reduction of tile qmap sum, deterministic partial per tile
  for (int o = 16; o > 0; o >>= 1) qsum += __shfl_xor(qsum, o, 32);
  if (lane == 0) qpart[(size_t)b * 1024 + by * 32 + bx] = qsum;
}

// ---------------- k3: Q[b] = mean of qmap ----------------
__global__ __launch_bounds__(256) void k3_q(const float* __restrict__ qpart,
                                            float* __restrict__ out)
{
  const int b = blockIdx.x;
  const int t = threadIdx.x;
  __shared__ float s[256];
  float v = qpart[(size_t)b * 1024 + t] +
            qpart[(size_t)b * 1024 + 256 + t] +
            qpart[(size_t)b * 1024 + 512 + t] +
            qpart[(size_t)b * 1024 + 768 + t];
  s[t] = v; __syncthreads();
  for (int r = 128; r > 0; r >>= 1) {
    if (t < r) s[t] += s[t + r];
    __syncthreads();
  }
  if (t == 0) out[OFF_Q + b] = s[0] * (1.0f / ((float)HH * (float)WW));
}

// ---------------- launch ----------------
extern "C" void kernel_launch(void* const* d_in, const int* in_sizes, int n_in,
                              void* d_out, int out_size, void* d_ws, size_t ws_size,
                              hipStream_t stream) {
  (void)in_sizes; (void)n_in; (void)out_size; (void)ws_size;
  const float* X = (const float*)d_in[0];   // LDRseq
  const float* Y = (const float*)d_in[1];   // Fused
  float* out = (float*)d_out;

  float* ws    = (float*)d_ws;
  float* hs    = ws;                                   // 11 hsum fields
  float* gpart = ws + HS_FLOATS;
  float* gmu   = gpart + GPART_FLOATS;
  float* qpart = gmu + GMU_FLOATS;

  k1_rowsum<<<dim3(HH, BB), 512, 0, stream>>>(X, Y, hs, gpart);
  kg_gmu<<<BB * NN, 128, 0, stream>>>(gpart, gmu);
  k2_wmma_ssim<<<dim3(WW / 16, HH / 16, BB), 32, 0, stream>>>(hs, gmu, out, qpart);
  k3_q<<<BB, 256, 0, stream>>>(qpart, out);
}